// CustomLinear_20160576488054
// MI455X (gfx1250) — compile-verified
//
#include <hip/hip_runtime.h>

typedef __attribute__((ext_vector_type(2))) float v2f;
typedef __attribute__((ext_vector_type(4))) float v4f;
typedef __attribute__((ext_vector_type(8))) float v8f;

#define NTOK   (8 * 4096)   // B*N tokens
#define EDIM   130          // E = 2*D + 2
#define DDIM   64
#define HE     (8 * EDIM)   // 1040 floats per token per tensor
#define ROWS   32           // tokens per block
#define ROWPAD 1044         // padded LDS row (floats); avoids half-wave bank conflicts

__global__ __launch_bounds__(256)
void qkv_wmma_kernel(const float* __restrict__ x,
                     const float* __restrict__ Mq, const float* __restrict__ Bq,
                     const float* __restrict__ Mk, const float* __restrict__ Bk,
                     const float* __restrict__ Mv,
                     float* __restrict__ out)
{
    __shared__ float lds[ROWS * ROWPAD];

    const int tid      = threadIdx.x;
    const int t        = blockIdx.y;            // 0=q, 1=k, 2=v (uniform)
    const int tok_base = blockIdx.x * ROWS;

    // ---- Phase A: zero the output slab in LDS (vectorized) ----
    {
        v4f z = {0.f, 0.f, 0.f, 0.f};
        v4f* l4 = (v4f*)lds;
        for (int i = tid; i < ROWS * ROWPAD / 4; i += 256) l4[i] = z;
    }
    __syncthreads();

    // Uniform per-tensor parameters
    const float* M       = (t == 0) ? Mq : ((t == 1) ? Mk : Mv);
    const int    src_off = (t == 0) ? (DDIM + 1) : 0;   // q reads x2, k/v read x1
    const int    nhloop  = (t == 2) ? 2 : 1;            // v: each wave does 2 heads

    const int wave = tid >> 5;
    const int lane = tid & 31;
    const int tile = wave >> 2;      // 0..1 : 16-token tile within block
    const int hrel = wave & 3;       // head (q/k) or head%4 (v)
    const int m    = lane & 15;      // token row within tile (A) / output col (B,D)
    const int kk   = lane >> 4;      // half-wave -> K offset of 2

    const float* xrow = x + (size_t)(tok_base + tile * 16 + m) * EDIM + src_off;

    // ---- Phase B: WMMA matvecs, D -> LDS ----
    for (int hh = 0; hh < nhloop; ++hh) {
        const int h = hrel + hh * 4;
        const float* Mh = M + (size_t)h * (DDIM * DDIM);

        v8f acc0 = {}, acc1 = {}, acc2 = {}, acc3 = {};
        #pragma unroll
        for (int kc = 0; kc < 16; ++kc) {
            const int j = kc * 4 + kk * 2;          // global K index (per ISA A/B layout)
            v2f a;                                   // x row: 4B-aligned -> scalar loads
            a.x = xrow[j];
            a.y = xrow[j + 1];
            const float* bp = Mh + m * DDIM + j;     // even index -> 8B aligned
            v2f b0 = *(const v2f*)(bp + 0 * 16 * DDIM);
            v2f b1 = *(const v2f*)(bp + 1 * 16 * DDIM);
            v2f b2 = *(const v2f*)(bp + 2 * 16 * DDIM);
            v2f b3 = *(const v2f*)(bp + 3 * 16 * DDIM);
            acc0 = __builtin_amdgcn_wmma_f32_16x16x4_f32(false, a, false, b0, (short)0, acc0, false, false);
            acc1 = __builtin_amdgcn_wmma_f32_16x16x4_f32(false, a, false, b1, (short)0, acc1, false, false);
            acc2 = __builtin_amdgcn_wmma_f32_16x16x4_f32(false, a, false, b2, (short)0, acc2, false, false);
            acc3 = __builtin_amdgcn_wmma_f32_16x16x4_f32(false, a, false, b3, (short)0, acc3, false, false);
        }

        // D layout: vgpr r, half-wave kk -> token (r + 8*kk), col n = m (ISA 7.12.2)
        const int colbase = h * EDIM + (DDIM + 1);
        float* ldrow = &lds[(tile * 16 + kk * 8) * ROWPAD + colbase + m];
        #pragma unroll
        for (int r = 0; r < 8; ++r) {
            ldrow[r * ROWPAD + 0 * 16] = acc0[r];
            ldrow[r * ROWPAD + 1 * 16] = acc1[r];
            ldrow[r * ROWPAD + 2 * 16] = acc2[r];
            ldrow[r * ROWPAD + 3 * 16] = acc3[r];
        }
    }

    // Bias terms (q/k only): 32 tokens x 8 heads = 256 scalar writes, one per thread.
    if (t < 2) {
        const int tokr = tid >> 3;
        const int idx  = tid & 7;
        const float* bias = (t == 0) ? Bq : Bk;
        const float* row  = x + (size_t)(tok_base + tokr) * EDIM;
        // q: always s_last; k: heads<4 use s_last, heads>=4 use s_mid
        float s = (t == 0 || idx < 4) ? row[2 * DDIM + 1] : row[DDIM];
        int col = (idx < 4) ? (idx * EDIM + 2 * DDIM + 1) : (idx * EDIM + DDIM + 1);
        lds[tokr * ROWPAD + col] = s * bias[idx];
    }
    __syncthreads();

    // ---- Phase C: coalesced non-temporal b128 streamout ----
    // 409 MB of write-once data: TH=NT keeps the 192MB L2 free for M/x reuse.
    float* ob = out + (size_t)t * NTOK * HE + (size_t)tok_base * HE;
    for (int i = tid; i < ROWS * (HE / 4); i += 256) {
        int tokr = i / (HE / 4);
        int c    = i - tokr * (HE / 4);
        v4f v = *(const v4f*)&lds[tokr * ROWPAD + c * 4];
        __builtin_nontemporal_store(v, (v4f*)&ob[(size_t)tokr * HE + c * 4]);
    }
}

extern "C" void kernel_launch(void* const* d_in, const int* in_sizes, int n_in,
                              void* d_out, int out_size, void* d_ws, size_t ws_size,
                              hipStream_t stream) {
    (void)in_sizes; (void)n_in; (void)out_size; (void)d_ws; (void)ws_size;
    const float* x  = (const float*)d_in[0];
    const float* Mq = (const float*)d_in[1];
    const float* Bq = (const float*)d_in[2];
    const float* Mk = (const float*)d_in[3];
    const float* Bk = (const float*)d_in[4];
    const float* Mv = (const float*)d_in[5];
    float* out = (float*)d_out;

    dim3 grid(NTOK / ROWS, 3, 1);   // 1024 token-blocks x {q,k,v}
    qkv_wmma_kernel<<<grid, 256, 0, stream>>>(x, Mq, Bq, Mk, Bk, Mv, out);
}